// CurriculumDMGHANmae_25993142075981
// MI455X (gfx1250) — compile-verified
//
#include <hip/hip_runtime.h>

typedef __attribute__((ext_vector_type(2))) float v2f;
typedef __attribute__((ext_vector_type(4))) float v4f;
typedef __attribute__((ext_vector_type(8))) float v8f;

#define LSEQ 12
#define LDSP 20   // LDS row stride (dwords): even (8B-aligned b64 frags) and
                  // conflict-free across both lane-halves (10x==1 mod 32 unsolvable)

__device__ __forceinline__ float sigmoidf_(float x) { return 1.f / (1.f + __expf(-x)); }

// ---------------------------------------------------------------------------
// Generic f32 WMMA GEMM:  Out[M,N] = act(A[M,K] (lda) @ W[N,K]^T (ldw) + bias[N]
//                                        + extra[(m/extraDiv)*extraLd + n])
// block = 256 threads = 8 waves; block tile 128(M) x 64(N).
// Each wave computes a 16x64 strip: 4 v8f accumulators, A-frag reused 4x.
// Requirements: K % 16 == 0, M % 128 == 0. N-edge guarded (N = 5..1500).
// act: 0=none 1=relu 2=sigmoid
// ---------------------------------------------------------------------------
__global__ __launch_bounds__(256) void gemm_wmma_f32(
    const float* __restrict__ A, int lda,
    const float* __restrict__ W, int ldw,
    const float* __restrict__ bias,
    const float* __restrict__ extra, int extraDiv, int extraLd,
    float* __restrict__ Out, int ldo,
    int M, int N, int K, int act)
{
    __shared__ float sA[128 * LDSP];
    __shared__ float sW[64 * LDSP];

    const int tid  = threadIdx.x;
    const int wave = tid >> 5;      // 0..7 -> M sub-tile
    const int lane = tid & 31;
    const int half = lane >> 4;     // 0: lanes 0-15, 1: lanes 16-31
    const int r    = lane & 15;
    const int mBlk = blockIdx.y * 128;
    const int nBlk = blockIdx.x * 64;

    // cooperative tile-load indices (coalesced b128)
    const int ar = tid >> 1;            // 0..127 A row
    const int ac = (tid & 1) << 3;      // 0 or 8 (two v4f per thread)
    const int wr = tid >> 2;            // 0..63  W row
    const int wc = (tid & 3) << 2;      // 0,4,8,12 (one v4f per thread)
    const bool wIn = (nBlk + wr) < N;

    const float* gA = A + (size_t)(mBlk + ar) * lda + ac;
    const float* gW = W + (size_t)(nBlk + wr) * ldw + wc;

    v8f acc[4];
#pragma unroll
    for (int j = 0; j < 4; ++j) acc[j] = (v8f){0.f,0.f,0.f,0.f,0.f,0.f,0.f,0.f};

    for (int k0 = 0; k0 < K; k0 += 16) {
        __syncthreads();
        {   // A tile 128x16
            v4f t0 = *reinterpret_cast<const v4f*>(gA + k0);
            v4f t1 = *reinterpret_cast<const v4f*>(gA + k0 + 4);
            if (k0 + 16 < K) __builtin_prefetch(gA + k0 + 16, 0, 1);
            float* s = &sA[ar * LDSP + ac];
            s[0] = t0.x; s[1] = t0.y; s[2] = t0.z; s[3] = t0.w;
            s[4] = t1.x; s[5] = t1.y; s[6] = t1.z; s[7] = t1.w;
        }
        {   // W tile 64x16 (zero-fill past N)
            v4f t = {0.f, 0.f, 0.f, 0.f};
            if (wIn) {
                t = *reinterpret_cast<const v4f*>(gW + k0);
                if (k0 + 16 < K) __builtin_prefetch(gW + k0 + 16, 0, 1);
            }
            float* s = &sW[wr * LDSP + wc];
            s[0] = t.x; s[1] = t.y; s[2] = t.z; s[3] = t.w;
        }
        __syncthreads();
#pragma unroll
        for (int kk = 0; kk < 16; kk += 4) {
            // A 16x4 frag: lanes 0-15 hold K={kk,kk+1}, lanes 16-31 K={kk+2,kk+3}
            v2f a = *reinterpret_cast<const v2f*>(&sA[(wave * 16 + r) * LDSP + kk + 2 * half]);
#pragma unroll
            for (int j = 0; j < 4; ++j) {
                v2f b = *reinterpret_cast<const v2f*>(&sW[(j * 16 + r) * LDSP + kk + 2 * half]);
                acc[j] = __builtin_amdgcn_wmma_f32_16x16x4_f32(false, a, false, b,
                                                               (short)0, acc[j], false, false);
            }
        }
    }

    // C/D layout: VGPR i -> M = i + 8*half ; N = lane%16
    const int gm0 = mBlk + wave * 16 + half * 8;
#pragma unroll
    for (int j = 0; j < 4; ++j) {
        const int gn = nBlk + j * 16 + r;
        if (gn < N) {
            const float bv = bias ? bias[gn] : 0.f;
#pragma unroll
            for (int i = 0; i < 8; ++i) {
                const int gm = gm0 + i;
                float v = acc[j][i] + bv;
                if (extra) v += extra[(size_t)(gm / extraDiv) * extraLd + gn];
                if (act == 1)      v = fmaxf(v, 0.f);
                else if (act == 2) v = sigmoidf_(v);
                Out[(size_t)gm * ldo + gn] = v;
            }
        }
    }
}

// ---------------------------------------------------------------------------
// In-place LayerNorm over dim 256; one wave32 per row, 8 rows per block.
// ---------------------------------------------------------------------------
__global__ __launch_bounds__(256) void layernorm_kernel(
    float* __restrict__ x, const float* __restrict__ g, const float* __restrict__ b)
{
    const int wave = threadIdx.x >> 5;
    const int lane = threadIdx.x & 31;
    const int row  = blockIdx.x * 8 + wave;
    float* xr = x + (size_t)row * 256;

    float v[8];
    float s = 0.f;
#pragma unroll
    for (int j = 0; j < 8; ++j) { v[j] = xr[lane + 32 * j]; s += v[j]; }
#pragma unroll
    for (int off = 16; off > 0; off >>= 1) s += __shfl_xor(s, off, 32);
    const float mean = s * (1.f / 256.f);

    float q = 0.f;
#pragma unroll
    for (int j = 0; j < 8; ++j) { const float d = v[j] - mean; q += d * d; }
#pragma unroll
    for (int off = 16; off > 0; off >>= 1) q += __shfl_xor(q, off, 32);
    const float inv = rsqrtf(q * (1.f / 256.f) + 1e-5f);

#pragma unroll
    for (int j = 0; j < 8; ++j) {
        const int c = lane + 32 * j;
        xr[c] = (v[j] - mean) * inv * g[c] + b[c];
    }
}

// ---------------------------------------------------------------------------
// Causal depthwise conv (D_CONV=4) + bias + SiLU on xc half of xz.
// xz: (B*12, 1024), xc_out: (B*12, 512)
// ---------------------------------------------------------------------------
__global__ void conv_silu_kernel(const float* __restrict__ xz,
                                 const float* __restrict__ cw,
                                 const float* __restrict__ cb,
                                 float* __restrict__ xc, int total)
{
    const int idx = blockIdx.x * blockDim.x + threadIdx.x;
    if (idx >= total) return;
    const int d   = idx & 511;
    const int row = idx >> 9;         // b*12 + l
    const int l   = row % LSEQ;
    const int b   = row / LSEQ;
    float acc = cb[d];
    const float* base = xz + (size_t)b * LSEQ * 1024 + d;
#pragma unroll
    for (int k = 0; k < 4; ++k) {
        const int ls = l + k - 3;
        if (ls >= 0) acc += base[(size_t)ls * 1024] * cw[d * 4 + k];
    }
    xc[idx] = acc * sigmoidf_(acc);   // SiLU
}

// ---------------------------------------------------------------------------
// Fused Mamba core: dt_proj(16->512)+softplus, selective scan (16 states, L=12),
// skip + gate with silu(z). One block (512 threads) per batch element.
// Writes y into the (dead) xc half of xz: xz[m*1024 + d].
// ---------------------------------------------------------------------------
__global__ __launch_bounds__(512) void mamba_scan_kernel(
    const float* __restrict__ dbc,    // (B*12, 48): [0:16]=dt_r, [16:32]=B, [32:48]=C
    const float* __restrict__ xc,     // (B*12, 512) post-conv SiLU
    float* __restrict__ xz,           // (B*12, 1024): z at +512; y written at +0
    const float* __restrict__ dtw,    // (512,16)
    const float* __restrict__ dtb,    // (512,)
    const float* __restrict__ Alog,   // (512,16)
    const float* __restrict__ Dp)     // (512,)
{
    __shared__ float s[LSEQ * 48];
    const int b = blockIdx.x;
    const int d = threadIdx.x;
    for (int i = d; i < LSEQ * 48; i += 512) s[i] = dbc[(size_t)b * LSEQ * 48 + i];
    __syncthreads();

    float w[16], a[16], h[16];
#pragma unroll
    for (int n = 0; n < 16; ++n) {
        w[n] = dtw[d * 16 + n];
        a[n] = -__expf(Alog[d * 16 + n]);
        h[n] = 0.f;
    }
    const float bdt = dtb[d];
    const float dpv = Dp[d];

    for (int t = 0; t < LSEQ; ++t) {
        const float* row = &s[t * 48];
        float pre = bdt;
#pragma unroll
        for (int n = 0; n < 16; ++n) pre += row[n] * w[n];
        const float dt = (pre > 20.f) ? pre : log1pf(__expf(pre));   // softplus

        const size_t m = (size_t)(b * LSEQ + t);
        const float xcv = xc[m * 512 + d];
        const float zv  = xz[m * 1024 + 512 + d];
        float y = 0.f;
#pragma unroll
        for (int n = 0; n < 16; ++n) {
            h[n] = __expf(dt * a[n]) * h[n] + dt * row[16 + n] * xcv;
            y += h[n] * row[32 + n];
        }
        xz[m * 1024 + d] = (y + xcv * dpv) * (zv * sigmoidf_(zv));
    }
}

// ---------------------------------------------------------------------------
// pooled[b, j] = mean_t seq[b, t, j]
// ---------------------------------------------------------------------------
__global__ void pool_kernel(const float* __restrict__ seq, float* __restrict__ pooled, int total)
{
    const int idx = blockIdx.x * blockDim.x + threadIdx.x;
    if (idx >= total) return;
    const int b = idx >> 8;
    const int j = idx & 255;
    float sum = 0.f;
#pragma unroll
    for (int t = 0; t < LSEQ; ++t) sum += seq[((size_t)b * LSEQ + t) * 256 + j];
    pooled[idx] = sum * (1.f / 12.f);
}

// ---------------------------------------------------------------------------
// S = G*P + (1-G)*Q
// ---------------------------------------------------------------------------
__global__ void gate_mix_kernel(const float* __restrict__ G, const float* __restrict__ P,
                                const float* __restrict__ Q, float* __restrict__ S, int n)
{
    const int i = blockIdx.x * blockDim.x + threadIdx.x;
    if (i >= n) return;
    const float g = G[i];
    S[i] = g * P[i] + (1.f - g) * Q[i];
}

// ---------------------------------------------------------------------------
extern "C" void kernel_launch(void* const* d_in, const int* in_sizes, int n_in,
                              void* d_out, int out_size, void* d_ws, size_t ws_size,
                              hipStream_t stream)
{
    const float* coi        = (const float*)d_in[0];
    const float* coi_mae    = (const float*)d_in[1];
    const float* coi_pw     = (const float*)d_in[2];
    const float* coi_pb     = (const float*)d_in[3];
    const float* mae_pw     = (const float*)d_in[4];
    const float* mae_pb     = (const float*)d_in[5];
    const float* fusion_w   = (const float*)d_in[6];
    const float* fusion_b   = (const float*)d_in[7];
    const float* ln_g       = (const float*)d_in[8];
    const float* ln_b       = (const float*)d_in[9];
    const float* in_proj_w  = (const float*)d_in[10];
    const float* conv_w     = (const float*)d_in[11];
    const float* conv_b     = (const float*)d_in[12];
    const float* x_proj_w   = (const float*)d_in[13];
    const float* dt_proj_w  = (const float*)d_in[14];
    const float* dt_proj_b  = (const float*)d_in[15];
    const float* A_log      = (const float*)d_in[16];
    const float* Dp         = (const float*)d_in[17];
    const float* out_proj_w = (const float*)d_in[18];
    const float* attn_in_w  = (const float*)d_in[19];
    const float* attn_in_b  = (const float*)d_in[20];
    const float* attn_out_w = (const float*)d_in[21];
    const float* attn_out_b = (const float*)d_in[22];
    const float* gate_w     = (const float*)d_in[23];
    const float* gate_b     = (const float*)d_in[24];

    static const int NCLS[6] = {5, 30, 80, 200, 600, 1500};
    const float* clsw[6];
    const float* clsb[6];
    if (n_in >= 37) {
        for (int j = 0; j < 6; ++j) { clsw[j] = (const float*)d_in[25 + j];
                                      clsb[j] = (const float*)d_in[31 + j]; }
    } else {  // tuples concatenated
        const float* wb = (const float*)d_in[25];
        const float* bb = (const float*)d_in[26];
        size_t wo = 0, bo = 0;
        for (int j = 0; j < 6; ++j) { clsw[j] = wb + wo; clsb[j] = bb + bo;
                                      wo += (size_t)NCLS[j] * 256; bo += NCLS[j]; }
    }

    const int B  = in_sizes[1] / 768;   // 2048
    const int Mr = B * LSEQ;            // 24576

    float* ws = (float*)d_ws;
    size_t off = 0;
    auto alloc = [&](size_t n) { float* p = ws + off; off += n; return p; };
    float* CM  = alloc((size_t)Mr * 256);    // coi_main
    float* MP  = alloc((size_t)B * 256);     // mae projected
    float* MC  = alloc((size_t)B * 256);     // mae contribution through fusion_w[:,256:]
    float* FU  = alloc((size_t)Mr * 256);    // fused (LN in place)
    float* XZ  = alloc((size_t)Mr * 1024);   // in_proj output; y overwrites xc half
    float* XCC = alloc((size_t)Mr * 512);    // conv+silu output
    float* DBC = alloc((size_t)Mr * 48);     // x_proj output
    float* SEQ = alloc((size_t)Mr * 256);    // out_proj output
    float* PO  = alloc((size_t)B * 256);     // pooled
    float* P   = alloc((size_t)B * 256);     // prev feature
    float* V   = alloc((size_t)B * 256);     // value proj
    float* G   = alloc((size_t)B * 256);     // gate
    float* S   = alloc((size_t)B * 256);     // mixed src
    float* T   = alloc((size_t)B * 256);     // pooled gate half
    (void)ws_size; (void)out_size;

    auto gemm = [&](const float* A, int lda, const float* W, int ldw, const float* bias,
                    const float* extra, int ediv, int eld,
                    float* O, int ldo, int M, int N, int K, int act) {
        dim3 grid((N + 63) / 64, M / 128);
        hipLaunchKernelGGL(gemm_wmma_f32, grid, dim3(256), 0, stream,
                           A, lda, W, ldw, bias, extra, ediv, eld, O, ldo, M, N, K, act);
    };
    const float* NUL = nullptr;

    // 1) projections
    gemm(coi, 768, coi_pw, 768, coi_pb, NUL, 1, 0, CM, 256, Mr, 256, 768, 0);
    gemm(coi_mae, 768, mae_pw, 768, mae_pb, NUL, 1, 0, MP, 256, B, 256, 768, 0);
    // mae half of fusion (fusion_w[:, 256:512]), no bias (bias added once below)
    gemm(MP, 256, fusion_w + 256, 512, NUL, NUL, 1, 0, MC, 256, B, 256, 256, 0);
    // fusion: relu(CM @ fusion_w[:, :256]^T + MC[b] + fusion_b)
    gemm(CM, 256, fusion_w, 512, fusion_b, MC, LSEQ, 256, FU, 256, Mr, 256, 256, 1);
    // 2) layernorm in place
    hipLaunchKernelGGL(layernorm_kernel, dim3(Mr / 8), dim3(256), 0, stream, FU, ln_g, ln_b);
    // 3) mamba
    gemm(FU, 256, in_proj_w, 256, NUL, NUL, 1, 0, XZ, 1024, Mr, 1024, 256, 0);
    {
        const int total = Mr * 512;
        hipLaunchKernelGGL(conv_silu_kernel, dim3((total + 255) / 256), dim3(256), 0, stream,
                           XZ, conv_w, conv_b, XCC, total);
    }
    gemm(XCC, 512, x_proj_w, 512, NUL, NUL, 1, 0, DBC, 48, Mr, 48, 512, 0);
    hipLaunchKernelGGL(mamba_scan_kernel, dim3(B), dim3(512), 0, stream,
                       DBC, XCC, XZ, dt_proj_w, dt_proj_b, A_log, Dp);
    gemm(XZ, 1024, out_proj_w, 512, NUL, NUL, 1, 0, SEQ, 256, Mr, 256, 512, 0);
    {
        const int total = B * 256;
        hipLaunchKernelGGL(pool_kernel, dim3((total + 255) / 256), dim3(256), 0, stream,
                           SEQ, PO, total);
    }

    // 4) hierarchical head
    size_t outOff = 0;
    for (int i = 0; i < 6; ++i) {
        const float* src;
        if (i == 0) {
            src = PO;
        } else {
            const float* gw = gate_w + (size_t)(i - 1) * 256 * 512;
            const float* gb = gate_b + (size_t)(i - 1) * 256;
            // T = pooled @ gw[:,256:]^T + gb
            gemm(PO, 256, gw + 256, 512, gb, NUL, 1, 0, T, 256, B, 256, 256, 0);
            // G = sigmoid(prev @ gw[:,:256]^T + T)
            gemm(P, 256, gw, 512, NUL, T, 1, 256, G, 256, B, 256, 256, 2);
            hipLaunchKernelGGL(gate_mix_kernel, dim3((B * 256 + 255) / 256), dim3(256), 0,
                               stream, G, P, PO, S, B * 256);
            src = S;
        }
        const float* wv = attn_in_w + ((size_t)i * 768 + 512) * 256;
        const float* bv = attn_in_b + (size_t)i * 768 + 512;
        gemm(src, 256, wv, 256, bv, NUL, 1, 0, V, 256, B, 256, 256, 0);
        gemm(V, 256, attn_out_w + (size_t)i * 256 * 256, 256, attn_out_b + (size_t)i * 256,
             NUL, 1, 0, P, 256, B, 256, 256, 0);
        // classifier -> d_out (flat, in return order)
        gemm(P, 256, clsw[i], 256, clsb[i], NUL, 1, 0,
             (float*)d_out + outOff, NCLS[i], B, NCLS[i], 256, 0);
        outOff += (size_t)B * NCLS[i];
    }
}